// SimplePointerGenerator_85014582657769
// MI455X (gfx1250) — compile-verified
//
#include <hip/hip_runtime.h>
#include <hip/hip_bf16.h>
#include <math.h>
#include <stdint.h>

typedef float v2f __attribute__((ext_vector_type(2)));
typedef float v8f __attribute__((ext_vector_type(8)));

#define B_  8
#define T_  256
#define H_  1024
#define S_  128
#define D_  130
#define V_  32000
#define M_  (B_ * T_)       // 2048 rows
#define OD  (V_ + D_)       // 32130 output row stride

// GEMM tiling
#define KC   32             // K-chunk per LDS stage
#define LDP  44             // LDS row pitch in floats (conflict-free, 16B aligned)

// ---------------------------------------------------------------------------
// CDNA5 async global->LDS helpers
// ---------------------------------------------------------------------------
__device__ __forceinline__ void async_b128_to_lds(const float* gptr, float* lds_generic) {
  // Low 32 bits of a generic LDS pointer are the wave-relative LDS byte address
  // (ISA: LDS aperture lives in addr[63:32]).
  uint32_t lds_off = (uint32_t)(uintptr_t)lds_generic;
  asm volatile("global_load_async_to_lds_b128 %0, %1, off"
               :: "v"(lds_off), "v"(gptr)
               : "memory");
}

__device__ __forceinline__ void wait_asynccnt0() {
#if __has_builtin(__builtin_amdgcn_s_wait_asynccnt)
  __builtin_amdgcn_s_wait_asynccnt(0);
#else
  asm volatile("s_wait_asynccnt 0x0" ::: "memory");
#endif
}

// ---------------------------------------------------------------------------
// Kernel 1: p_copy[m] = sigmoid(hiddens[m,:] . w_ptr + b_ptr)
// ---------------------------------------------------------------------------
__global__ __launch_bounds__(256) void pcopy_kernel(
    const float* __restrict__ hiddens, const float* __restrict__ w_ptr,
    const float* __restrict__ b_ptr, float* __restrict__ pc) {
  __shared__ float red[256];
  const int m = blockIdx.x, tid = threadIdx.x;
  const float4 h = *(const float4*)(hiddens + (size_t)m * H_ + tid * 4);
  const float4 w = *(const float4*)(w_ptr + tid * 4);
  red[tid] = h.x * w.x + h.y * w.y + h.z * w.z + h.w * w.w;
  __syncthreads();
  for (int off = 128; off > 0; off >>= 1) {
    if (tid < off) red[tid] += red[tid + off];
    __syncthreads();
  }
  if (tid == 0) {
    float x = red[0] + b_ptr[0];
    pc[m] = 1.0f / (1.0f + __expf(-x));
  }
}

// ---------------------------------------------------------------------------
// Kernel 2: scores = hiddens @ W^T + b  via V_WMMA_F32_16X16X4_F32
//   8 waves as 2(M) x 4(N); wave tile 32x64 (8 accum tiles);
//   WG tile 64M x 256N; KC=32 double-buffered LDS fed by async-to-LDS DMA.
// ---------------------------------------------------------------------------
__device__ __forceinline__ void issue_chunk(
    const float* __restrict__ A, const float* __restrict__ W,
    int mBlock, int nBlock, int kc,
    float* __restrict__ sA, float* __restrict__ sB, int tid) {
  // A chunk: 64 rows x 32 floats = 512 float4 -> 2 per thread
#pragma unroll
  for (int i = 0; i < 2; ++i) {
    const int idx = tid + i * 256;
    const int r = idx >> 3, k4 = idx & 7;
    async_b128_to_lds(A + (size_t)(mBlock + r) * H_ + kc + k4 * 4,
                      sA + r * LDP + k4 * 4);
  }
  // B chunk: 256 rows x 32 floats = 2048 float4 -> 8 per thread
#pragma unroll
  for (int i = 0; i < 8; ++i) {
    const int idx = tid + i * 256;
    const int n = idx >> 3, k4 = idx & 7;
    async_b128_to_lds(W + (size_t)(nBlock + n) * H_ + kc + k4 * 4,
                      sB + n * LDP + k4 * 4);
  }
}

__global__ __launch_bounds__(256) void gemm_kernel(
    const float* __restrict__ A,     // hiddens [M_, H_]
    const float* __restrict__ W,     // [V_, H_]
    const float* __restrict__ bias,  // [V_]
    float* __restrict__ out) {       // [M_, OD], cols 0..V_-1
  __shared__ float sA[2][64 * LDP];   //  2 x 11264 B
  __shared__ float sB[2][256 * LDP];  //  2 x 45056 B  (total ~110 KB)
  const int tid  = threadIdx.x;
  const int lane = tid & 31;
  const int wv   = tid >> 5;        // 0..7
  const int wm   = wv >> 2;         // 0..1  (M waves)
  const int wn   = wv & 3;          // 0..3  (N waves)
  const int half = lane >> 4;       // 0..1
  const int lrow = lane & 15;       // 0..15
  const int mBlock = blockIdx.y * 64;
  const int nBlock = blockIdx.x * 256;

  const v8f zero = {0.f, 0.f, 0.f, 0.f, 0.f, 0.f, 0.f, 0.f};
  v8f acc[8] = {zero, zero, zero, zero, zero, zero, zero, zero};

  issue_chunk(A, W, mBlock, nBlock, 0, sA[0], sB[0], tid);

  int cur = 0;
  for (int kc = 0; kc < H_; kc += KC) {
    wait_asynccnt0();
    __syncthreads();                      // all waves' async data landed
    if (kc + KC < H_)
      issue_chunk(A, W, mBlock, nBlock, kc + KC, sA[cur ^ 1], sB[cur ^ 1], tid);

    const float* cA = sA[cur];
    const float* cB = sB[cur];
    const int aRow0 = (wm * 32 + lrow) * LDP + 2 * half;
    const int aRow1 = (wm * 32 + 16 + lrow) * LDP + 2 * half;
    const int bBase = 2 * half;

    // Software-pipelined fragment loads: prefetch k+4 while issuing WMMA on k.
    v2f aC[2], bC[4], aN[2], bN[4];
    aC[0] = *(const v2f*)&cA[aRow0];
    aC[1] = *(const v2f*)&cA[aRow1];
#pragma unroll
    for (int t = 0; t < 4; ++t)
      bC[t] = *(const v2f*)&cB[(wn * 64 + t * 16 + lrow) * LDP + bBase];

#pragma unroll
    for (int k = 0; k < KC; k += 4) {
      if (k + 4 < KC) {
        aN[0] = *(const v2f*)&cA[aRow0 + k + 4];
        aN[1] = *(const v2f*)&cA[aRow1 + k + 4];
#pragma unroll
        for (int t = 0; t < 4; ++t)
          bN[t] = *(const v2f*)&cB[(wn * 64 + t * 16 + lrow) * LDP + bBase + k + 4];
      }
#pragma unroll
      for (int mi = 0; mi < 2; ++mi)
#pragma unroll
        for (int t = 0; t < 4; ++t)
          acc[mi * 4 + t] = __builtin_amdgcn_wmma_f32_16x16x4_f32(
              false, aC[mi], false, bC[t], (short)0, acc[mi * 4 + t],
              false, false);
      if (k + 4 < KC) {
        aC[0] = aN[0]; aC[1] = aN[1];
#pragma unroll
        for (int t = 0; t < 4; ++t) bC[t] = bN[t];
      }
    }
    cur ^= 1;
  }

  // D layout: VGPR r -> M = r (lanes 0-15) / r+8 (lanes 16-31), N = lane&15.
#pragma unroll
  for (int mi = 0; mi < 2; ++mi) {
    const int m0 = mBlock + wm * 32 + mi * 16 + half * 8;
#pragma unroll
    for (int t = 0; t < 4; ++t) {
      const int col = nBlock + wn * 64 + t * 16 + lrow;
      const float bb = bias[col];
#pragma unroll
      for (int r = 0; r < 8; ++r)
        out[(size_t)(m0 + r) * OD + col] = acc[mi * 4 + t][r] + bb;
    }
  }
}

// ---------------------------------------------------------------------------
// Kernel 3: online softmax stats per row over cols 1..V_-1 (col 0 = -inf).
// ---------------------------------------------------------------------------
__global__ __launch_bounds__(256) void softstat_kernel(
    const float* __restrict__ out, float* __restrict__ ws) {
  __shared__ float smax[256];
  __shared__ float ssum[256];
  const int m = blockIdx.x, tid = threadIdx.x;
  const float* row = out + (size_t)m * OD;
  float mx = -INFINITY, sm = 0.f;
  for (int v = 1 + tid; v < V_; v += 256) {
    float x = row[v];
    if (x > mx) { sm = sm * __expf(mx - x) + 1.0f; mx = x; }
    else        { sm += __expf(x - mx); }
  }
  smax[tid] = mx; ssum[tid] = sm;
  __syncthreads();
  for (int off = 128; off > 0; off >>= 1) {
    if (tid < off) {
      float m1 = smax[tid], s1 = ssum[tid];
      float m2 = smax[tid + off], s2 = ssum[tid + off];
      float M = fmaxf(m1, m2);
      smax[tid] = M;
      ssum[tid] = s1 * __expf(m1 - M) + s2 * __expf(m2 - M);
    }
    __syncthreads();
  }
  if (tid == 0) { ws[M_ + m] = smax[0]; ws[2 * M_ + m] = ssum[0]; }
}

// ---------------------------------------------------------------------------
// Kernel 4: finalize probs (vocab + copy sections), argmax -> predictions.
// ---------------------------------------------------------------------------
__global__ __launch_bounds__(256) void finalize_kernel(
    const float* __restrict__ att,   // [B,T,S]
    const float* __restrict__ amap,  // [B,S,D]
    const float* __restrict__ ws,
    float* __restrict__ out,         // probs [M_, OD]
    float* __restrict__ preds) {     // [M_]
  __shared__ float bval[256];
  __shared__ int   bidx[256];
  const int m = blockIdx.x, tid = threadIdx.x;
  const float pc = ws[m], rmax = ws[M_ + m];
  const float rinv = 1.0f / ws[2 * M_ + m];
  const float pgen = 1.0f - pc;
  float* row = out + (size_t)m * OD;
  float best = -1.0f; int bi = 0;

  for (int v = tid; v < V_; v += 256) {
    float p = (v == 0) ? 0.0f : __expf(row[v] - rmax) * rinv * pgen;
    row[v] = p;
    if (p > best) { best = p; bi = v; }
  }
  // copy distribution: row[V_+d] = pc * sum_s att[m,s] * amap[b,s,d]
  const int b = m / T_;
  if (tid < D_) {
    const float* attrow = att + (size_t)m * S_;
    const float* mapb = amap + (size_t)b * S_ * D_;
    float s = 0.f;
    for (int si = 0; si < S_; ++si) s += attrow[si] * mapb[si * D_ + tid];
    float p = s * pc;
    row[V_ + tid] = p;
    if (p > best) { best = p; bi = V_ + tid; }
  }
  bval[tid] = best; bidx[tid] = bi;
  __syncthreads();
  for (int off = 128; off > 0; off >>= 1) {
    if (tid < off && bval[tid + off] > bval[tid]) {
      bval[tid] = bval[tid + off]; bidx[tid] = bidx[tid + off];
    }
    __syncthreads();
  }
  if (tid == 0) preds[m] = (float)bidx[0];
}

// ---------------------------------------------------------------------------
extern "C" void kernel_launch(void* const* d_in, const int* in_sizes, int n_in,
                              void* d_out, int out_size, void* d_ws, size_t ws_size,
                              hipStream_t stream) {
  const float* hiddens = (const float*)d_in[0];  // [B,T,H]
  const float* att     = (const float*)d_in[1];  // [B,T,S]
  const float* amap    = (const float*)d_in[2];  // [B,S,D]
  const float* W       = (const float*)d_in[3];  // [V,H]
  const float* bias    = (const float*)d_in[4];  // [V]
  const float* w_ptr   = (const float*)d_in[5];  // [H,1]
  const float* b_ptr   = (const float*)d_in[6];  // [1]

  float* probs = (float*)d_out;                  // [M_, OD]
  float* preds = probs + (size_t)M_ * OD;        // [M_]
  float* ws    = (float*)d_ws;                   // 3*M_ floats

  pcopy_kernel<<<M_, 256, 0, stream>>>(hiddens, w_ptr, b_ptr, ws);

  dim3 g(V_ / 256, M_ / 64);                     // (125, 32)
  gemm_kernel<<<g, 256, 0, stream>>>(hiddens, W, bias, probs);

  softstat_kernel<<<M_, 256, 0, stream>>>(probs, ws);
  finalize_kernel<<<M_, 256, 0, stream>>>(att, amap, ws, probs, preds);
}